// GE2ELoss_24266565223033
// MI455X (gfx1250) — compile-verified
//
#include <hip/hip_runtime.h>
#include <math.h>

// ---------------- problem constants (from reference) ----------------
#define N_SPK   1024
#define M_UTT   32
#define M_ENR   16
#define D_DIM   512
#define N_TEST  (N_SPK * M_ENR)      // 16384 test vectors
#define EPS_F   1e-8f

// GEMM tiling: each wave handles a 16-row tile x 128-column chunk (8 WMMA tiles)
#define COLS_PER_CHUNK 128
#define CHUNKS (N_TEST / COLS_PER_CHUNK)   // 128

typedef float v2f __attribute__((ext_vector_type(2)));
typedef float v8f __attribute__((ext_vector_type(8)));

// -------------------------------------------------------------------
// Kernel 1: centroid of first 16 utterances per speaker, L2-normalize.
// -------------------------------------------------------------------
__global__ void __launch_bounds__(128)
ge2e_centroid_kernel(const float* __restrict__ emb, float* __restrict__ cent_n) {
    const int spk = blockIdx.x;
    const int tid = threadIdx.x;
    const float* base = emb + (size_t)spk * M_UTT * D_DIM;

    float c[4];
    float ss = 0.0f;
#pragma unroll
    for (int j = 0; j < 4; ++j) {
        const int d = tid + j * 128;
        float s = 0.0f;
#pragma unroll
        for (int u = 0; u < M_ENR; ++u)
            s += base[(size_t)u * D_DIM + d];
        s *= (1.0f / (float)M_ENR);
        c[j] = s;
        ss += s * s;
    }
    for (int m = 16; m >= 1; m >>= 1) ss += __shfl_xor(ss, m, 32);
    __shared__ float red[4];
    const int lane = tid & 31, wid = tid >> 5;
    if (lane == 0) red[wid] = ss;
    __syncthreads();
    const float tot = red[0] + red[1] + red[2] + red[3];
    const float scale = 1.0f / fmaxf(sqrtf(tot), EPS_F);
#pragma unroll
    for (int j = 0; j < 4; ++j) {
        const int d = tid + j * 128;
        cent_n[(size_t)spk * D_DIM + d] = c[j] * scale;
    }
}

// -------------------------------------------------------------------
// Kernel 2: L2-normalize each test embedding (utterances 16..31).
// -------------------------------------------------------------------
__global__ void __launch_bounds__(128)
ge2e_testnorm_kernel(const float* __restrict__ emb, float* __restrict__ test_n) {
    const int r   = blockIdx.x;          // flat test index
    const int spk = r >> 4;
    const int t   = r & 15;
    const int tid = threadIdx.x;
    const float* src = emb + ((size_t)spk * M_UTT + M_ENR + t) * D_DIM;

    float v[4];
    float ss = 0.0f;
#pragma unroll
    for (int j = 0; j < 4; ++j) {
        const int d = tid + j * 128;
        v[j] = src[d];
        ss += v[j] * v[j];
    }
    for (int m = 16; m >= 1; m >>= 1) ss += __shfl_xor(ss, m, 32);
    __shared__ float red[4];
    const int lane = tid & 31, wid = tid >> 5;
    if (lane == 0) red[wid] = ss;
    __syncthreads();
    const float tot = red[0] + red[1] + red[2] + red[3];
    const float scale = 1.0f / fmaxf(sqrtf(tot), EPS_F);
#pragma unroll
    for (int j = 0; j < 4; ++j) {
        const int d = tid + j * 128;
        test_n[(size_t)r * D_DIM + d] = v[j] * scale;
    }
}

// -------------------------------------------------------------------
// Kernel 3: S = alpha * cent_n @ test_n^T + beta via V_WMMA_F32_16X16X4_F32,
// fused exp + row-sum (total and diagonal/positive) per (rowTile, chunk).
//
// Double-buffered K loop: the 9 loads (1 A + 8 B float2 fragments) for step
// k+4 are issued before the 8 WMMAs consuming step k, so the hardware wait
// becomes loadcnt<=9 instead of loadcnt==0 (L2 latency hidden behind the
// matrix pipe). All 8 B tiles are addressed from one base pointer with
// immediate offsets (t * 16*512*4 = t*32768 bytes, within 24-bit IOFFSET).
// -------------------------------------------------------------------
__global__ void __launch_bounds__(256)
ge2e_gemm_kernel(const float* __restrict__ cent_n,
                 const float* __restrict__ test_n,
                 const float* __restrict__ alphaPtr,
                 const float* __restrict__ betaPtr,
                 float* __restrict__ tot_part,
                 float* __restrict__ pos_part) {
    const int rowBase = blockIdx.x * 16;
    const int wid     = threadIdx.x >> 5;
    const int lane    = threadIdx.x & 31;
    const int chunk   = blockIdx.y * 8 + wid;
    const int colChunkBase = chunk * COLS_PER_CHUNK;

    const int row16 = lane & 15;
    const int half  = lane >> 4;
    const int koff  = half * 2;

    const float alpha = alphaPtr[0];
    const float beta  = betaPtr[0];

    const float* aBase = cent_n + (size_t)(rowBase + row16) * D_DIM + koff;
    const float* bBase = test_n + (size_t)(colChunkBase + row16) * D_DIM + koff;

    v8f acc[8];
#pragma unroll
    for (int t = 0; t < 8; ++t) acc[t] = (v8f){0,0,0,0,0,0,0,0};

#define LOAD_STEP(kk, aReg, bReg)                                              \
    do {                                                                       \
        aReg = *(const v2f*)(aBase + (kk));                                    \
        _Pragma("unroll")                                                      \
        for (int t = 0; t < 8; ++t)                                            \
            bReg[t] = *(const v2f*)(bBase + (size_t)t * 16 * D_DIM + (kk));    \
    } while (0)

#define WMMA_STEP(aReg, bReg)                                                  \
    do {                                                                       \
        _Pragma("unroll")                                                      \
        for (int t = 0; t < 8; ++t)                                            \
            acc[t] = __builtin_amdgcn_wmma_f32_16x16x4_f32(                    \
                false, aReg, false, bReg[t], (short)0, acc[t], false, false);  \
    } while (0)

    v2f a0, a1;
    v2f b0[8], b1[8];
    LOAD_STEP(0, a0, b0);
    for (int k = 0; k < D_DIM; k += 8) {
        LOAD_STEP(k + 4, a1, b1);          // prefetch step k+4
        WMMA_STEP(a0, b0);                 // consume step k
        const int kn = (k + 8 < D_DIM) ? (k + 8) : 0;   // dummy tail reload
        LOAD_STEP(kn, a0, b0);             // prefetch step k+8
        WMMA_STEP(a1, b1);                 // consume step k+4
    }
#undef LOAD_STEP
#undef WMMA_STEP

    // exp(alpha*s + beta); accumulate per-lane totals and diagonal positives.
    v8f tot8 = (v8f){0,0,0,0,0,0,0,0};
    v8f pos8 = (v8f){0,0,0,0,0,0,0,0};
#pragma unroll
    for (int t = 0; t < 8; ++t) {
        const int jSpk = (colChunkBase + t * 16) >> 4;   // speaker of this col tile
        const int dr   = jSpk - rowBase;                 // diagonal row within tile
        const bool diagHalf = (dr >= 0) && (dr < 16) && ((dr >> 3) == half);
        const int  dvr = dr & 7;
#pragma unroll
        for (int vr = 0; vr < 8; ++vr) {
            const float e = __expf(alpha * acc[t][vr] + beta);
            tot8[vr] += e;
            pos8[vr] += (diagHalf && (vr == dvr)) ? e : 0.0f;  // branch-free select
        }
    }

    // butterfly reduce within each 16-lane group (columns of the tile)
#pragma unroll
    for (int vr = 0; vr < 8; ++vr) {
        float tsum = tot8[vr];
        float psum = pos8[vr];
        for (int m = 8; m >= 1; m >>= 1) {
            tsum += __shfl_xor(tsum, m, 16);
            psum += __shfl_xor(psum, m, 16);
        }
        if (row16 == 0) {
            const int row = rowBase + half * 8 + vr;
            tot_part[(size_t)chunk * N_SPK + row] = tsum;
            pos_part[(size_t)chunk * N_SPK + row] = psum;
        }
    }
}

// -------------------------------------------------------------------
// Kernel 4: per-speaker reduce over chunks, loss = mean(log(neg)-log(pos)).
// One block of 256 threads; fixed-order -> deterministic.
// -------------------------------------------------------------------
__global__ void __launch_bounds__(256)
ge2e_finalize_kernel(const float* __restrict__ tot_part,
                     const float* __restrict__ pos_part,
                     float* __restrict__ out) {
    const int tid = threadIdx.x;
    float lsum = 0.0f;
    for (int r = tid; r < N_SPK; r += 256) {
        float tot = 0.0f, pos = 0.0f;
        for (int c = 0; c < CHUNKS; ++c) {
            tot += tot_part[(size_t)c * N_SPK + r];
            pos += pos_part[(size_t)c * N_SPK + r];
        }
        const float neg = tot - pos;
        lsum += logf(neg) - logf(pos);
    }
    __shared__ float sdata[256];
    sdata[tid] = lsum;
    __syncthreads();
    for (int s = 128; s >= 1; s >>= 1) {
        if (tid < s) sdata[tid] += sdata[tid + s];
        __syncthreads();
    }
    if (tid == 0) out[0] = sdata[0] / (float)N_SPK;
}

// -------------------------------------------------------------------
extern "C" void kernel_launch(void* const* d_in, const int* in_sizes, int n_in,
                              void* d_out, int out_size, void* d_ws, size_t ws_size,
                              hipStream_t stream) {
    const float* emb      = (const float*)d_in[0];
    // d_in[1] = labels (int64): structurally arange-repeat, unused (matches ref math)
    const float* alphaPtr = (const float*)d_in[2];
    const float* betaPtr  = (const float*)d_in[3];
    float* out = (float*)d_out;

    // workspace layout (floats)
    float* ws      = (float*)d_ws;
    float* cent_n  = ws;                                   // 1024*512      = 524288
    float* test_n  = cent_n + (size_t)N_SPK * D_DIM;       // 16384*512     = 8388608
    float* tot_prt = test_n + (size_t)N_TEST * D_DIM;      // 128*1024      = 131072
    float* pos_prt = tot_prt + (size_t)CHUNKS * N_SPK;     // 128*1024      = 131072
    (void)in_sizes; (void)n_in; (void)out_size; (void)ws_size;

    ge2e_centroid_kernel<<<dim3(N_SPK), dim3(128), 0, stream>>>(emb, cent_n);
    ge2e_testnorm_kernel<<<dim3(N_TEST), dim3(128), 0, stream>>>(emb, test_n);
    ge2e_gemm_kernel<<<dim3(N_SPK / 16, CHUNKS / 8), dim3(256), 0, stream>>>(
        cent_n, test_n, alphaPtr, betaPtr, tot_prt, pos_prt);
    ge2e_finalize_kernel<<<dim3(1), dim3(256), 0, stream>>>(tot_prt, pos_prt, out);
}